// Recommender_22505628631474
// MI455X (gfx1250) — compile-verified
//
#include <hip/hip_runtime.h>
#include <math.h>

#define NU 20000
#define NI 20000
#define NT 5000
#define NN 45000
#define EMBD 64
#define NEDGE 645000
#define STRIDE 68   // padded LDS row stride (floats): 16B-aligned, bank-conflict-free

typedef float v2f __attribute__((ext_vector_type(2)));
typedef float v8f __attribute__((ext_vector_type(8)));

// ---------------- float atomic max via int/uint ordering trick ----------------
__device__ __forceinline__ void atomic_max_f32(float* addr, float val) {
  if (val >= 0.0f) atomicMax((int*)addr, __float_as_int(val));
  else             atomicMin((unsigned int*)addr, __float_as_uint(val));
}

// ---------------- fused edge MLP (two 64x64 GEMMs via fp32 WMMA) --------------
// PASS 0: segMax[head][d] = atomic-max of logits
// PASS 1: den[head][d] += exp(a-m);  num[head][d] += exp(a-m)*emb_e[d]
template <int PASS>
__global__ __launch_bounds__(64) void edge_mlp_kernel(
    const float* __restrict__ allE,
    const float* __restrict__ W1, const float* __restrict__ b1,
    const float* __restrict__ W2, const float* __restrict__ b2,
    const int* __restrict__ head, const int* __restrict__ tail,
    float* __restrict__ segMax, float* __restrict__ numAcc,
    float* __restrict__ denAcc, int nEdges) {
  __shared__ __align__(16) float sW1[64 * STRIDE];
  __shared__ __align__(16) float sW2[64 * STRIDE];
  __shared__ float sB1[64];
  __shared__ float sB2[64];
  __shared__ __align__(16) float sEmb[2][16 * STRIDE];
  __shared__ __align__(16) float sH[2][16 * STRIDE];
  __shared__ int sHead[2][16];

  const int tid  = threadIdx.x;
  const int wave = tid >> 5;   // 0..1
  const int lane = tid & 31;
  const int half = lane >> 4;  // 0/1: which K/M half this lane serves
  const int l15  = lane & 15;

  // stage weights (padded rows)
  for (int i = tid; i < 64 * 64; i += 64) {
    int r = i >> 6, c = i & 63;
    sW1[r * STRIDE + c] = W1[i];
    sW2[r * STRIDE + c] = W2[i];
  }
  if (tid < 64) { sB1[tid] = b1[tid]; sB2[tid] = b2[tid]; }

  const int tileBase = (blockIdx.x * 2 + wave) * 16;

  if (lane < 16) {
    int e = tileBase + lane;
    sHead[wave][lane] = head[e < nEdges ? e : (nEdges - 1)];
  }
  {  // gather 16 emb rows: 2 rows / iteration, float4 per lane
    const int quad = lane & 15;
    const int sub  = lane >> 4;
    for (int t = 0; t < 8; ++t) {
      int r = t * 2 + sub;
      int e = tileBase + r;
      int trow = tail[e < nEdges ? e : (nEdges - 1)];
      const float4 v = *(const float4*)(allE + (size_t)trow * EMBD + quad * 4);
      *(float4*)(&sEmb[wave][r * STRIDE + quad * 4]) = v;
    }
  }
  __syncthreads();

  // ---- GEMM1: h = relu(emb @ W1^T + b1) -> sH ----
  // A 16x4 f32 layout: lane (half,l15) holds row l15, K = 4k+2*half .. +1
  // B 4x16 f32 layout: lane (half,l15) holds col n=nt*16+l15, same K pair
  for (int nt = 0; nt < 4; ++nt) {
    const int n = nt * 16 + l15;
    v8f acc = {};
#pragma unroll
    for (int k = 0; k < 16; ++k) {
      const int kk = 4 * k + 2 * half;
      v2f a = *(const v2f*)(&sEmb[wave][l15 * STRIDE + kk]);
      v2f b = *(const v2f*)(&sW1[n * STRIDE + kk]);
      acc = __builtin_amdgcn_wmma_f32_16x16x4_f32(false, a, false, b,
                                                  (short)0, acc, false, false);
    }
    const float bias = sB1[n];
#pragma unroll
    for (int j = 0; j < 8; ++j) {
      float hv = acc[j] + bias;                       // C: VGPR j = row j+8*half
      sH[wave][(j + 8 * half) * STRIDE + n] = hv > 0.0f ? hv : 0.0f;
    }
  }
  __syncthreads();

  // ---- GEMM2: a = h @ W2^T + b2 (kept in registers, C layout) ----
  v8f accA[4];
  for (int nt = 0; nt < 4; ++nt) {
    const int n = nt * 16 + l15;
    v8f acc = {};
#pragma unroll
    for (int k = 0; k < 16; ++k) {
      const int kk = 4 * k + 2 * half;
      v2f a = *(const v2f*)(&sH[wave][l15 * STRIDE + kk]);
      v2f b = *(const v2f*)(&sW2[n * STRIDE + kk]);
      acc = __builtin_amdgcn_wmma_f32_16x16x4_f32(false, a, false, b,
                                                  (short)0, acc, false, false);
    }
    const float bias = sB2[n];
#pragma unroll
    for (int j = 0; j < 8; ++j) acc[j] += bias;
    accA[nt] = acc;
  }

  // ---- epilogue: scatter per (edge m, feature n) ----
  for (int nt = 0; nt < 4; ++nt) {
    const int n = nt * 16 + l15;
#pragma unroll
    for (int j = 0; j < 8; ++j) {
      const int m = j + 8 * half;
      const int e = tileBase + m;
      if (e < nEdges) {
        const size_t addr = (size_t)sHead[wave][m] * EMBD + n;
        const float av = accA[nt][j];
        if (PASS == 0) {
          atomic_max_f32(segMax + addr, av);
        } else {
          const float w = __expf(av - segMax[addr]);
          atomicAdd(denAcc + addr, w);
          atomicAdd(numAcc + addr, w * sEmb[wave][m * STRIDE + n]);
        }
      }
    }
  }
}

// ---------------- scratch fills ----------------
__global__ void fill_kernel(unsigned* p, unsigned bits, int n) {
  int i = blockIdx.x * blockDim.x + threadIdx.x;
  if (i < n) p[i] = bits;
}

// ---------------- build allE / allO from inputs ----------------
__global__ void init_in_kernel(const float* __restrict__ ue, const float* __restrict__ uo,
                               const float* __restrict__ ie, const float* __restrict__ io,
                               const float* __restrict__ te, const float* __restrict__ to,
                               float* __restrict__ allE, float* __restrict__ allO) {
  int idx = blockIdx.x * blockDim.x + threadIdx.x;
  if (idx >= NN * EMBD) return;
  int node = idx >> 6;
  float e, o;
  if (node < NU)            { e = ue[idx];                       o = uo[idx]; }
  else if (node < NU + NI)  { e = ie[idx - NU * EMBD];           o = io[idx - NU * EMBD]; }
  else                      { e = te[idx - (NU + NI) * EMBD];    o = to[idx - (NU + NI) * EMBD]; }
  allE[idx] = e;
  allO[idx] = fmaxf(o, 0.0f);
}

// ---------------- box-offset edge pass (masked segment min/max) ----------------
__global__ void off_edge_kernel(const float* __restrict__ allO,
                                const int* __restrict__ head, const int* __restrict__ tail,
                                unsigned* __restrict__ iu, unsigned* __restrict__ ut,
                                unsigned* __restrict__ itm, unsigned* __restrict__ tg,
                                int nEdges) {
  int e = blockIdx.x * blockDim.x + threadIdx.x;
  if (e >= nEdges) return;
  const int h = head[e], t = tail[e];
  const float* src = allO + (size_t)t * EMBD;
  unsigned* dst; bool isMin;
  if (h < NU) {
    if (t < NU) return;                                      // user<-user: in no mask
    if (t < NU + NI) { dst = iu + (size_t)h * EMBD;  isMin = true; }
    else             { dst = ut + (size_t)h * EMBD;  isMin = false; }
  } else if (h < NU + NI) { dst = itm + (size_t)(h - NU) * EMBD;        isMin = false; }
  else                    { dst = tg  + (size_t)(h - NU - NI) * EMBD;   isMin = true;  }
  for (int d = 0; d < EMBD; ++d) {
    float v = fmaxf(src[d], 0.0f);        // relu'd: non-negative -> uint-ordered
    unsigned b = __float_as_uint(v);
    if (isMin) atomicMin(dst + d, b);
    else       atomicMax(dst + d, b);
  }
}

// ---------------- agg = num/den, then L2 row-normalize (1 wave / row) ----------
__global__ __launch_bounds__(256) void agg_norm_kernel(const float* __restrict__ num,
                                                       const float* __restrict__ den,
                                                       float* __restrict__ allE, int nNodes) {
  int w = (blockIdx.x * blockDim.x + threadIdx.x) >> 5;
  int lane = threadIdx.x & 31;
  if (w >= nNodes) return;
  const size_t base = (size_t)w * EMBD;
  float v0 = num[base + lane]      / den[base + lane];
  float v1 = num[base + lane + 32] / den[base + lane + 32];
  float ss = v0 * v0 + v1 * v1;
  for (int off = 16; off > 0; off >>= 1) ss += __shfl_xor(ss, off, 32);
  const float scale = 1.0f / fmaxf(sqrtf(ss), 1e-12f);
  allE[base + lane]      = v0 * scale;
  allE[base + lane + 32] = v1 * scale;
}

// ---------------- merge offset reductions into allO ----------------
__global__ void off_final_kernel(const unsigned* __restrict__ iu, const unsigned* __restrict__ ut,
                                 const unsigned* __restrict__ itm, const unsigned* __restrict__ tg,
                                 float* __restrict__ allO) {
  int idx = blockIdx.x * blockDim.x + threadIdx.x;
  if (idx >= NN * EMBD) return;
  int node = idx >> 6;
  float v;
  if (node < NU) {
    unsigned ab = iu[idx];
    float a = (ab == 0x7F800000u) ? 0.0f : __uint_as_float(ab);  // empty min -> 0
    float b = __uint_as_float(ut[idx]);                          // max init 0 already correct
    v = fminf(a, b);
  } else if (node < NU + NI) {
    v = __uint_as_float(itm[idx - NU * EMBD]);
  } else {
    unsigned ab = tg[idx - (NU + NI) * EMBD];
    v = (ab == 0x7F800000u) ? 0.0f : __uint_as_float(ab);
  }
  allO[idx] = fmaxf(v, 0.0f);
}

// ---------------- reorder into tuple output layout ----------------
__global__ void out_kernel(const float* __restrict__ allE, const float* __restrict__ allO,
                           float* __restrict__ out) {
  int idx = blockIdx.x * blockDim.x + threadIdx.x;
  const int NUe = NU * EMBD, NIe = NI * EMBD, NTe = NT * EMBD;
  if (idx >= 2 * (NUe + NIe + NTe)) return;
  if (idx < NUe)                           out[idx] = allE[idx];
  else if (idx < 2 * NUe)                  out[idx] = allO[idx - NUe];
  else if (idx < 2 * NUe + NIe)            out[idx] = allE[NUe + idx - 2 * NUe];
  else if (idx < 2 * NUe + 2 * NIe)        out[idx] = allO[NUe + idx - (2 * NUe + NIe)];
  else if (idx < 2 * NUe + 2 * NIe + NTe)  out[idx] = allE[NUe + NIe + idx - (2 * NUe + 2 * NIe)];
  else                                     out[idx] = allO[NUe + NIe + idx - (2 * NUe + 2 * NIe + NTe)];
}

extern "C" void kernel_launch(void* const* d_in, const int* in_sizes, int n_in,
                              void* d_out, int out_size, void* d_ws, size_t ws_size,
                              hipStream_t stream) {
  const float* ue = (const float*)d_in[0];
  const float* uo = (const float*)d_in[1];
  const float* ie = (const float*)d_in[2];
  const float* io = (const float*)d_in[3];
  const float* te = (const float*)d_in[4];
  const float* to = (const float*)d_in[5];
  const float* W1 = (const float*)d_in[6];
  const float* b1 = (const float*)d_in[7];
  const float* W2 = (const float*)d_in[8];
  const float* b2 = (const float*)d_in[9];
  const int* head = (const int*)d_in[10];
  const int* tail = (const int*)d_in[11];

  const size_t NNE = (size_t)NN * EMBD;  // 2,880,000
  float* ws     = (float*)d_ws;
  float* allE   = ws;
  float* allO   = ws + 1 * NNE;
  float* segMax = ws + 2 * NNE;
  float* numA   = ws + 3 * NNE;
  float* denA   = ws + 4 * NNE;
  unsigned* iu  = (unsigned*)(ws + 5 * NNE);
  unsigned* ut  = iu  + (size_t)NU * EMBD;
  unsigned* itm = ut  + (size_t)NU * EMBD;
  unsigned* tg  = itm + (size_t)NI * EMBD;

  const int BLK = 256;
  const int gNN   = (int)((NNE + BLK - 1) / BLK);
  const int gE    = (NEDGE + BLK - 1) / BLK;
  const int gOut  = (int)((2 * NNE + BLK - 1) / BLK);
  const int nTiles  = (NEDGE + 15) / 16;
  const int gTiles  = (nTiles + 1) / 2;  // 2 waves (tiles) per 64-thread block
  const int gWaves  = (NN * 32 + BLK - 1) / BLK;

  init_in_kernel<<<gNN, BLK, 0, stream>>>(ue, uo, ie, io, te, to, allE, allO);

  for (int hop = 0; hop < 2; ++hop) {
    fill_kernel<<<gNN, BLK, 0, stream>>>((unsigned*)segMax, 0xFF800000u, (int)NNE); // -inf
    fill_kernel<<<gNN, BLK, 0, stream>>>((unsigned*)numA, 0u, (int)NNE);
    fill_kernel<<<gNN, BLK, 0, stream>>>((unsigned*)denA, 0u, (int)NNE);
    fill_kernel<<<(NU * EMBD + BLK - 1) / BLK, BLK, 0, stream>>>(iu, 0x7F800000u, NU * EMBD);
    fill_kernel<<<(NU * EMBD + BLK - 1) / BLK, BLK, 0, stream>>>(ut, 0u, NU * EMBD);
    fill_kernel<<<(NI * EMBD + BLK - 1) / BLK, BLK, 0, stream>>>(itm, 0u, NI * EMBD);
    fill_kernel<<<(NT * EMBD + BLK - 1) / BLK, BLK, 0, stream>>>(tg, 0x7F800000u, NT * EMBD);

    edge_mlp_kernel<0><<<gTiles, 64, 0, stream>>>(allE, W1, b1, W2, b2, head, tail,
                                                  segMax, numA, denA, NEDGE);
    edge_mlp_kernel<1><<<gTiles, 64, 0, stream>>>(allE, W1, b1, W2, b2, head, tail,
                                                  segMax, numA, denA, NEDGE);
    off_edge_kernel<<<gE, BLK, 0, stream>>>(allO, head, tail, iu, ut, itm, tg, NEDGE);
    agg_norm_kernel<<<gWaves, BLK, 0, stream>>>(numA, denA, allE, NN);
    off_final_kernel<<<gNN, BLK, 0, stream>>>(iu, ut, itm, tg, allO);
  }

  out_kernel<<<gOut, BLK, 0, stream>>>(allE, allO, (float*)d_out);
}